// LayerModule_18734647345218
// MI455X (gfx1250) — compile-verified
//
#include <hip/hip_runtime.h>

typedef __attribute__((ext_vector_type(2))) float v2f;
typedef __attribute__((ext_vector_type(8))) float v8f;
typedef __attribute__((ext_vector_type(4))) unsigned int u32x4;
typedef __attribute__((ext_vector_type(8))) int i32x8;
typedef __attribute__((ext_vector_type(4))) int i32x4;

#define GDIM 513
#define GG   (GDIM * GDIM)      // 263169 columns
#define WGN  128                // columns per workgroup (8 N-tiles of 16)
#define LDS_ROW    136          // 128 data DWORDs + 8 pad DWORDs (bank-conflict-free)
#define LDS_TILE   (64 * LDS_ROW)   // one source tile, in DWORDs (34816 B)

// ---------------------------------------------------------------------------
// Prep kernel: materialize the stacked 128x128 weight matrix W2 into scratch.
//   rows  o<64  : combined_under ->  [ w_neg | w_pos ]
//   rows  o>=64 : combined_over  ->  [ w_pos | w_neg ]
// ---------------------------------------------------------------------------
__global__ __launch_bounds__(256) void build_w2(const float* __restrict__ W,
                                                float* __restrict__ W2) {
    int idx = blockIdx.x * 256 + threadIdx.x;   // 0..16383
    int o = idx >> 7;
    int i = idx & 127;
    int oo = (o < 64) ? o : (o - 64);
    int ii = (i < 64) ? i : (i - 64);
    float w = W[oo * 64 + ii];
    bool use_pos = ((o < 64) != (i < 64));
    W2[idx] = use_pos ? fmaxf(w, 0.0f) : fminf(w, 0.0f);
}

// Issue one 2D TDM load: 64 rows x 128 cols of f32, row stride GG elements,
// zero-fill past 'tensor_cols', LDS rows padded to 136 DWORDs.
__device__ __forceinline__ void tdm_load_tile(const float* src, unsigned lds_byte_addr,
                                              unsigned tensor_cols) {
    unsigned long long ga = (unsigned long long)(uintptr_t)src;

    u32x4 g0;
    g0[0] = 1u;                                            // count=1, user mode
    g0[1] = lds_byte_addr;                                 // lds_addr
    g0[2] = (unsigned)(ga & 0xffffffffu);                  // global_addr[31:0]
    g0[3] = (unsigned)((ga >> 32) & 0x01ffffffu) | (2u << 30); // addr[56:32] | type=2

    i32x8 g1;
    g1[0] = (2 << 16) | (1 << 20) | (6 << 22) | (7 << 25); // 4B elems, pad: every 128dw +8dw
    g1[1] = (int)((tensor_cols & 0xffffu) << 16);          // tensor_dim0[15:0]
    g1[2] = (int)(((tensor_cols >> 16) & 0xffffu) | (64u << 16)); // dim0[31:16] | dim1=64
    g1[3] = (int)(128u << 16);                             // dim1[31:16]=0 | tile_dim0=128
    g1[4] = (int)64;                                       // tile_dim1=64, tile_dim2=0
    g1[5] = (int)(unsigned)GG;                             // tensor_dim0_stride[31:0]
    g1[6] = 0;                                             // stride[47:32]=0 | dim1_stride lo
    g1[7] = 0;

    i32x4 gz4 = {0, 0, 0, 0};                              // 2D tensor: groups 2/3 unused
    i32x8 gz8 = {0, 0, 0, 0, 0, 0, 0, 0};
    __builtin_amdgcn_tensor_load_to_lds(g0, g1, gz4, gz4, gz8, 0);
}

// ---------------------------------------------------------------------------
// Main kernel: D(128 x GG) = W2(128x128) @ [over; under](128 x GG) + bias
// TDM stages the 128x128-column B tile into LDS once per block; 8 waves each
// own one 16-row M-tile and run 2x ds_load + v_wmma_f32_16x16x4_f32 per K-step.
// ---------------------------------------------------------------------------
__global__ __launch_bounds__(256, 2) void relu_split_gemm_wmma(
    const float* __restrict__ in_under,   // (64, 513, 513)
    const float* __restrict__ in_over,    // (64, 513, 513)
    const float* __restrict__ W2,         // (128, 128) prebuilt in scratch
    const float* __restrict__ bias,       // (64,)
    float* __restrict__ out)              // (128, 513, 513): under rows then over rows
{
    __shared__ float btile[2 * LDS_TILE];   // over tile @ rows 0-63, under @ rows 64-127

    const int lane  = threadIdx.x & 31;
    const int wave  = threadIdx.x >> 5;   // 0..7 -> M-tile index
    const int half  = lane >> 4;
    const int n     = lane & 15;
    const int mbase = wave * 16;
    const int colbase0 = blockIdx.x * WGN;

    // Kick off the TDM loads first so they overlap the A/bias prologue.
    if (threadIdx.x < 32) {
        unsigned lds0 = (unsigned)(uintptr_t)&btile[0];
        unsigned cols_rem = (unsigned)(GG - colbase0);
        tdm_load_tile(in_over  + colbase0, lds0,                    cols_rem);
        tdm_load_tile(in_under + colbase0, lds0 + LDS_TILE * 4u,    cols_rem);
    }

    // A fragments: lane m holds K = 2*half + {0,1} -> adjacent in row-major W2.
    const v2f* __restrict__ w2row = (const v2f*)(W2 + (size_t)(mbase + n) * 128);
    v2f a[32];
#pragma unroll
    for (int kt = 0; kt < 32; ++kt)
        a[kt] = w2row[2 * kt + half];

    float bv[8];
#pragma unroll
    for (int r = 0; r < 8; ++r)
        bv[r] = bias[(mbase + r + 8 * half) & 63];

    if (threadIdx.x < 32)
        __builtin_amdgcn_s_wait_tensorcnt(0);
    __syncthreads();

#pragma unroll 1
    for (int nt = 0; nt < WGN / 16; ++nt) {
        const int colIdx = nt * 16 + n;          // 0..127 within the LDS tile
        const int col    = colbase0 + colIdx;

        v8f acc = {};

#pragma unroll
        for (int kt = 0; kt < 32; ++kt) {
            // Combined row kk = 4kt + 2*half (over rows 0-63, under rows 64-127
            // sit contiguously at kk*LDS_ROW thanks to the back-to-back tiles).
            const float* p = &btile[(4 * kt + 2 * half) * LDS_ROW + colIdx];
            v2f b;
            b.x = p[0];          // K = 2*half
            b.y = p[LDS_ROW];    // K = 2*half + 1  (544 B apart -> ds_load_2addr-able)
            acc = __builtin_amdgcn_wmma_f32_16x16x4_f32(
                false, a[kt], false, b, (short)0, acc, false, false);
        }

        if (col < GG) {
#pragma unroll
            for (int r = 0; r < 8; ++r) {
                int m = mbase + r + 8 * half;   // 0..127: <64 -> under block, >=64 -> over
                out[(size_t)m * GG + (size_t)col] = acc[r] + bv[r];
            }
        }
    }
}

extern "C" void kernel_launch(void* const* d_in, const int* in_sizes, int n_in,
                              void* d_out, int out_size, void* d_ws, size_t ws_size,
                              hipStream_t stream) {
    const float* in_under = (const float*)d_in[0];
    const float* in_over  = (const float*)d_in[1];
    const float* W        = (const float*)d_in[2];
    const float* bias     = (const float*)d_in[3];
    // d_in[4] = layer_idx (unused by the math)
    float* out = (float*)d_out;
    float* W2  = (float*)d_ws;   // 128*128*4 = 64 KB of scratch

    build_w2<<<(128 * 128) / 256, 256, 0, stream>>>(W, W2);

    const int blocks = (GG + WGN - 1) / WGN;   // 2057
    relu_split_gemm_wmma<<<blocks, 256, 0, stream>>>(in_under, in_over, W2, bias, out);
}